// IRadon_49967649521725
// MI455X (gfx1250) — compile-verified
//
#include <hip/hip_runtime.h>
#include <cstdint>

typedef __attribute__((ext_vector_type(2))) float v2f;
typedef __attribute__((ext_vector_type(8))) float v8f;

// Problem constants (x: (4,1,512,180) f32, out: (4,1,512,512) f32)
#define NB      4
#define HROWS   512           // sinogram detector rows
#define LANG    180           // real angles
#define CH      16            // angles per chunk (matches WMMA N)
#define NCHUNK  12            // 12*16 = 192 padded angles
#define WOUT    512           // output H=W
#define SINO_STRIDE (HROWS * LANG)   // elements per image

__global__ __launch_bounds__(256) void iradon_bp_kernel(
    const float* __restrict__ x, float* __restrict__ out)
{
    // Double-buffered sinogram chunk: [row 0..511][angle 0..15], 2 x 32KB
    __shared__ float lbuf[2][HROWS * CH];

    const int tid   = threadIdx.x;
    const int lane  = tid & 31;
    const int nlo   = lane & 15;          // angle slot within chunk / pixel half-select
    const int wv    = tid >> 5;           // wave id in block (0..7)
    const int n_img = blockIdx.z;
    const int i_row = blockIdx.y * 8 + wv;    // output row handled by this wave
    const int j0    = blockIdx.x * 16;        // output column tile base

    const uint64_t srcbase =
        (uint64_t)(uintptr_t)(x + (size_t)n_img * SINO_STRIDE);
    const uint32_t ldsbase = (uint32_t)(uintptr_t)(&lbuf[0][0]);

    // ---- async global->LDS transposing stage of one 16-angle chunk --------
    // Per chunk: 512 rows x 4 quad-angles of B128 = 2048 lane-tasks,
    // 8 async instructions per thread (ASYNCcnt += 8 per wave).
    auto issue_chunk = [&](int c, int cb) {
        const int l0 = c * CH;
        #pragma unroll
        for (int t = 0; t < 8; ++t) {
            const int k = tid + t * 256;          // 0..2047
            const int r = k >> 2;                 // sinogram row
            const int q = k & 3;                  // quad of 4 angles
            int ls = l0 + 4 * q;                  // source angle (16B aligned)
            if (ls > LANG - 4) ls = LANG - 4;     // clamp padded quads in-bounds
            const uint32_t goff = (uint32_t)((r * LANG + ls) * 4);
            const uint32_t dsb  =
                ldsbase + (uint32_t)(((cb * (HROWS * CH)) + r * CH + 4 * q) * 4);
            asm volatile("global_load_async_to_lds_b128 %0, %1, %2"
                         :: "v"(dsb), "v"(goff), "s"(srcbase) : "memory");
        }
    };

    // ---- WMMA A operand: pixel-side matrix, constant over chunks ----------
    // A(16x4): A[m,0]=u_{j0+m}, A[m,1]=u_i, A[m,2]=1, A[m,3]=0.
    // Lane<16 holds K={0,1}; lane>=16 holds K={2,3} (per CDNA5 16x4 f32 layout).
    const float u_i = fmaf((float)i_row,      2.0f / 511.0f, -1.0f);
    const float u_j = fmaf((float)(j0 + nlo), 2.0f / 511.0f, -1.0f);
    v2f amat;
    amat[0] = (lane < 16) ? u_j : 1.0f;
    amat[1] = (lane < 16) ? u_i : 0.0f;

    float acc[8] = {0.f, 0.f, 0.f, 0.f, 0.f, 0.f, 0.f, 0.f};

    issue_chunk(0, 0);

    for (int c = 0; c < NCHUNK; ++c) {
        const int cb = c & 1;
        if (c + 1 < NCHUNK) {
            issue_chunk(c + 1, cb ^ 1);
            asm volatile("s_wait_asynccnt 0x8" ::: "memory"); // chunk c done
        } else {
            asm volatile("s_wait_asynccnt 0x0" ::: "memory");
        }
        __syncthreads();   // chunk c visible to all waves

        // ---- WMMA B operand: angle-side matrix for this chunk ------------
        // B(4x16): B[0,n]=a*cos, B[1,n]=-a*sin, B[2,n]=bias_n, B[3,n]=0.
        // Lane<16 holds K={0,1}; lane>=16 holds K={2,3}.
        // Padded angles (l>=180) get bias=-3e8 -> iy<<0 -> both corners
        // invalid -> zero contribution (matches grid_sample zero padding).
        const int   la  = c * CH + nlo;
        const float rad = (float)la * 0.017453292519943295f;
        float sn, cs;
        __sincosf(rad, &sn, &cs);
        v2f bmat;
        if (lane < 16) { bmat[0] = 255.5f * cs;  bmat[1] = -255.5f * sn; }
        else           { bmat[0] = (la < LANG) ? 255.5f : -3.0e8f; bmat[1] = 0.0f; }

        v8f cz = {};
        // D[m,n] = iy for pixel (i_row, j0+m) at angle (c*16+n)
        v8f d = __builtin_amdgcn_wmma_f32_16x16x4_f32(
            false, amat, false, bmat, (short)0, cz, false, false);

        const float* __restrict__ sb = &lbuf[cb][0];
        #pragma unroll
        for (int v = 0; v < 8; ++v) {
            // lane<16: pixel j0+v ; lane>=16: pixel j0+v+8 ; angle col = nlo
            const float dc = d[v];
            const float f  = floorf(dc);
            float w1 = dc - f;
            float w0 = 1.0f - w1;
            const float f0 = fminf(fmaxf(f,        0.0f), 511.0f);
            const float f1 = fminf(fmaxf(f + 1.0f, 0.0f), 511.0f);
            const int   i0 = (int)f0;
            const int   i1 = (int)f1;
            const float s0 = sb[i0 * CH + nlo];
            const float s1 = sb[i1 * CH + nlo];
            w0 = (f >=  0.0f && f <= 511.0f) ? w0 : 0.0f;   // corner-0 valid
            w1 = (f >= -1.0f && f <= 510.0f) ? w1 : 0.0f;   // corner-1 valid
            acc[v] = fmaf(w0, s0, acc[v]);
            acc[v] = fmaf(w1, s1, acc[v]);
        }
        __syncthreads();   // all waves done reading before buffer reuse
    }

    // ---- reduce over the 16 angle-lanes sharing each pixel ---------------
    #pragma unroll
    for (int v = 0; v < 8; ++v) {
        #pragma unroll
        for (int ofs = 8; ofs >= 1; ofs >>= 1)
            acc[v] += __shfl_xor(acc[v], ofs, 32);
    }

    const float scale = 0.008726646259971648f;  // pi / 360
    if ((lane & 15) == 0) {
        const int jb = j0 + ((lane >> 4) << 3);  // lane0 -> j0..j0+7, lane16 -> +8
        float* op = out + ((size_t)n_img * WOUT + i_row) * WOUT + jb;
        float4 p0 = make_float4(acc[0] * scale, acc[1] * scale,
                                acc[2] * scale, acc[3] * scale);
        float4 p1 = make_float4(acc[4] * scale, acc[5] * scale,
                                acc[6] * scale, acc[7] * scale);
        *(float4*)(op)     = p0;
        *(float4*)(op + 4) = p1;
    }
}

extern "C" void kernel_launch(void* const* d_in, const int* in_sizes, int n_in,
                              void* d_out, int out_size, void* d_ws, size_t ws_size,
                              hipStream_t stream) {
    const float* x = (const float*)d_in[0];
    float* out = (float*)d_out;
    dim3 grid(WOUT / 16, WOUT / 8, NB);   // (32, 64, 4)
    dim3 block(256);                      // 8 waves: 8 rows x 16 cols per block
    hipLaunchKernelGGL(iradon_bp_kernel, grid, block, 0, stream, x, out);
}